// RiemannianAutoencoder_55061480735279
// MI455X (gfx1250) — compile-verified
//
#include <hip/hip_runtime.h>
#include <math.h>

// ---- problem constants (from the reference) ----
#define NB      50000
#define TSTEPS  100
#define LBOUND  4.5f                       // c * x_extent = 1.5 * 3.0
#define WBASE   (3.14159265358979f / (2.0f * LBOUND))
#define HSTEP   (1.0f / 99.0f)

typedef __attribute__((ext_vector_type(2))) float v2f;
typedef __attribute__((ext_vector_type(8))) float v8f;

// D(16x16 f32) = A(16x4 f32) * B(4x16 f32) + C  — CDNA5 WMMA, wave32
static __device__ __forceinline__ v8f wmma_f32(v2f a, v2f b, v8f c) {
  return __builtin_amdgcn_wmma_f32_16x16x4_f32(
      /*neg_a=*/false, a, /*neg_b=*/false, b,
      /*c_mod=*/(short)0, c, /*reuse_a=*/false, /*reuse_b=*/false);
}

// exchange with lane^16 (SWAPX16): group-of-32 ds_swizzle, xor=0x10, and=0x1f
static __device__ __forceinline__ float swap16(float x) {
  int i = __float_as_int(x);
  i = __builtin_amdgcn_ds_swizzle(i, 0x401F);
  return __int_as_float(i);
}

__global__ __launch_bounds__(256) void geodesic_wmma_kernel(
    const float* __restrict__ x0in, const float* __restrict__ v0in,
    const float* __restrict__ betain, const float* __restrict__ lsin,
    const float* __restrict__ alphain, float* __restrict__ out)
{
  const int lane = threadIdx.x & 31;
  const int row  = lane & 15;          // particle slot in wave; also m1 row of A operand
  const int hh   = lane >> 4;          // half id: 0 -> D rows 0..7, 1 -> D rows 8..15
  const float hf = (float)hh;          // 0.0f / 1.0f rotation selector (no cndmask chains)
  const int gw   = blockIdx.x * 8 + (threadIdx.x >> 5);
  const int b    = gw * 16 + row;      // global particle index
  const bool valid = (b < NB);
  const int  bc  = valid ? b : (NB - 1);

  const float ls0 = lsin[0], ls1 = lsin[1];
  const float alpha = alphain[0];
  const float spre = alpha * alpha * 6.28318530717958f * ls0 * ls1; // (2pi)^{n/2}, n=2

  // ---- constant A operands: Asq_d[m1,m2] = beta[d, m1*12+m2]*sqrt(S)*L^{-1}, 0-padded ----
  // A 16x4 f32 layout: lanes 0-15 hold M=row, VGPR0/1 = K-cols {4kk+2h, 4kk+2h+1}
  v2f Amat[3][3];
#pragma unroll
  for (int d = 0; d < 3; ++d) {
#pragma unroll
    for (int kk = 0; kk < 3; ++kk) {
      v2f a;
#pragma unroll
      for (int q = 0; q < 2; ++q) {
        int col = 4 * kk + 2 * hh + q;           // m2 in 0..11
        float val = 0.0f;
        if (row < 12) {
          float w0 = WBASE * (float)(row + 1);
          float w1 = WBASE * (float)(col + 1);
          float S  = spre * __expf(-0.5f * (ls0 * ls0 * w0 * w0 + ls1 * ls1 * w1 * w1));
          val = betain[d * 144 + row * 12 + col] * sqrtf(S) * (1.0f / LBOUND);
        }
        a[q] = val;
      }
      Amat[d][kk] = a;
    }
  }

  // ---- loop-invariant w-prefactors (already shifted for this lane half) ----
  float wrow[8];                       // for wcloc: WBASE*(r+1 + 8*hh)
#pragma unroll
  for (int r = 0; r < 8; ++r) wrow[r] = WBASE * (float)(r + 1) + hf * (8.0f * WBASE);
  float wcol[3][2];                    // for bwc2: WBASE*(4kk+q+1 + 2*hh)
#pragma unroll
  for (int kk = 0; kk < 3; ++kk)
#pragma unroll
    for (int q = 0; q < 2; ++q)
      wcol[kk][q] = WBASE * (float)(4 * kk + q + 1) + hf * (2.0f * WBASE);

  // ---- state (replicated in lanes p and p+16) ----
  float xx0 = x0in[bc * 2 + 0], xx1 = x0in[bc * 2 + 1];
  float vv0 = v0in[bc * 2 + 0], vv1 = v0in[bc * 2 + 1];
  float hdx0[4] = {0,0,0,0}, hdx1[4] = {0,0,0,0};   // x-derivative (=v) history, newest first
  float hdv0[4] = {0,0,0,0}, hdv1[4] = {0,0,0,0};   // v-derivative (=acc) history

  // t = 0 row: lane half 0 stores x-pair, half 1 stores v-pair  (out[t][c][b][i])
  if (valid) {
    size_t off = ((size_t)(0 * 2 + hh) * NB + (size_t)b) * 2;
    float2 w = hh ? make_float2(vv0, vv1) : make_float2(xx0, xx1);
    *reinterpret_cast<float2*>(out + off) = w;
  }

  for (int it = 0; it < TSTEPS - 1; ++it) {
    // ================= dim 0 (rows): s/c(k*th0), base k=1..8, half-1 rotated by 8*th0 ====
    float th0 = WBASE * (xx0 + LBOUND);
    float sA1 = __sinf(th0), cA1 = __cosf(th0);
    float skA[8], ckA[8];              // index m holds k=m+1
    skA[0] = sA1; ckA[0] = cA1;
#pragma unroll
    for (int m = 1; m < 8; ++m) {
      skA[m] = skA[m-1] * cA1 + ckA[m-1] * sA1;
      ckA[m] = ckA[m-1] * cA1 - skA[m-1] * sA1;
    }
    float soA = hf * skA[7];                       // hh ? sin(8*th0) : 0
    float coA = 1.0f + hf * (ckA[7] - 1.0f);       // hh ? cos(8*th0) : 1
    float s1loc[8], wcloc[8];
#pragma unroll
    for (int r = 0; r < 8; ++r) {                  // rows 12..15 multiply U==0: garbage OK
      s1loc[r]   = skA[r] * coA + ckA[r] * soA;
      float crot = ckA[r] * coA - skA[r] * soA;
      wcloc[r]   = wrow[r] * crot;
    }

    // ================= dim 1 (cols): base k in {1,2,5,6,9,10}, half-1 rotated by 2*th1 ===
    float th1 = WBASE * (xx1 + LBOUND);
    float sB1 = __sinf(th1), cB1 = __cosf(th1);
    float skB[10], ckB[10];            // index m holds k=m+1 (need up to k=10)
    skB[0] = sB1; ckB[0] = cB1;
#pragma unroll
    for (int m = 1; m < 10; ++m) {
      skB[m] = skB[m-1] * cB1 + ckB[m-1] * sB1;
      ckB[m] = ckB[m-1] * cB1 - skB[m-1] * sB1;
    }
    float soB = hf * skB[1];                       // hh ? sin(2*th1) : 0
    float coB = 1.0f + hf * (ckB[1] - 1.0f);       // hh ? cos(2*th1) : 1

    // B operands (4x16): lane holds column=particle, K rows {4kk+2h, 4kk+2h+1}
    v2f bs2[3], bwc2[3];
#pragma unroll
    for (int kk = 0; kk < 3; ++kk) {
      v2f p, q2;
#pragma unroll
      for (int q = 0; q < 2; ++q) {
        const int kb = 4 * kk + q;                 // base k-1 (constant index)
        float sv = skB[kb] * coB + ckB[kb] * soB;
        float cv = ckB[kb] * coB - skB[kb] * soB;
        p[q]  = sv;
        q2[q] = wcol[kk][q] * cv;
      }
      bs2[kk] = p; bwc2[kk] = q2;
    }

    // ---- 18 x v_wmma_f32_16x16x4_f32: U_d = A_d*s2, V_d = A_d*(w.c2) ----
    v8f U[3], V[3];
#pragma unroll
    for (int d = 0; d < 3; ++d) {
      v8f u = {0,0,0,0,0,0,0,0}, v = {0,0,0,0,0,0,0,0};
#pragma unroll
      for (int kk = 0; kk < 3; ++kk) {             // K = 12 -> 3 chunks of 4 (chunk 3 is 0)
        u = wmma_f32(Amat[d][kk], bs2[kk], u);
        v = wmma_f32(Amat[d][kk], bwc2[kk], v);
      }
      U[d] = u; V[d] = v;
    }

    // ---- f[d] = s1.U_d, d0f[d] = (w.c1).U_d, d1f[d] = s1.V_d  (half-dots + swap16) ----
    float f0, f1, f2, d0f0, d0f1, d0f2, d1f0, d1f1, d1f2;
    {
      float fr[3], g0[3], g1[3];
#pragma unroll
      for (int d = 0; d < 3; ++d) {
        float fp = 0.0f, dp0 = 0.0f, dp1 = 0.0f;
#pragma unroll
        for (int r = 0; r < 8; ++r) {
          fp  += s1loc[r] * U[d][r];
          dp0 += wcloc[r] * U[d][r];
          dp1 += s1loc[r] * V[d][r];
        }
        fr[d] = fp + swap16(fp);
        g0[d] = dp0 + swap16(dp0);
        g1[d] = dp1 + swap16(dp1);
      }
      f0 = fr[0]; f1 = fr[1]; f2 = fr[2];
      d0f0 = g0[0]; d0f1 = g0[1]; d0f2 = g0[2];
      d1f0 = g1[0]; d1f1 = g1[1]; d1f2 = g1[2];
    }

    // ---- 2x2 metric inverse + Christoffel contraction ----
    float g00 = 1.0f + f0, g01 = f1, g11 = 1.0f + f2;
    float inv = 1.0f / (g00 * g11 - g01 * g01);
    float gi00 = g11 * inv, gi01 = -g01 * inv, gi11 = g00 * inv;
    // A_l = sum_ij d_i g_{lj} v_i v_j ;  B_l = sum_ij d_l g_{ij} v_i v_j ;  q_l = 2A_l - B_l
    float Aq0 = vv0 * (d0f0 * vv0 + d0f1 * vv1) + vv1 * (d1f0 * vv0 + d1f1 * vv1);
    float Aq1 = vv0 * (d0f1 * vv0 + d0f2 * vv1) + vv1 * (d1f1 * vv0 + d1f2 * vv1);
    float Bq0 = d0f0 * vv0 * vv0 + 2.0f * d0f1 * vv0 * vv1 + d0f2 * vv1 * vv1;
    float Bq1 = d1f0 * vv0 * vv0 + 2.0f * d1f1 * vv0 * vv1 + d1f2 * vv1 * vv1;
    float q0 = 2.0f * Aq0 - Bq0, q1 = 2.0f * Aq1 - Bq1;
    float a0 = -0.5f * (gi00 * q0 + gi01 * q1);
    float a1 = -0.5f * (gi01 * q0 + gi11 * q1);

    // ---- Adams-Bashforth (order ramp 1..4), hist newest-first ----
#pragma unroll
    for (int j = 3; j > 0; --j) {
      hdx0[j] = hdx0[j-1]; hdx1[j] = hdx1[j-1];
      hdv0[j] = hdv0[j-1]; hdv1[j] = hdv1[j-1];
    }
    hdx0[0] = vv0; hdx1[0] = vv1; hdv0[0] = a0; hdv1[0] = a1;

    float c0a, c1a, c2a, c3a;
    if (it >= 3)      { c0a = 55.f/24.f; c1a = -59.f/24.f; c2a = 37.f/24.f; c3a = -9.f/24.f; }
    else if (it == 2) { c0a = 23.f/12.f; c1a = -16.f/12.f; c2a =  5.f/12.f; c3a = 0.f; }
    else if (it == 1) { c0a = 1.5f;      c1a = -0.5f;      c2a = 0.f;       c3a = 0.f; }
    else              { c0a = 1.0f;      c1a = 0.f;        c2a = 0.f;       c3a = 0.f; }

    xx0 += HSTEP * (c0a * hdx0[0] + c1a * hdx0[1] + c2a * hdx0[2] + c3a * hdx0[3]);
    xx1 += HSTEP * (c0a * hdx1[0] + c1a * hdx1[1] + c2a * hdx1[2] + c3a * hdx1[3]);
    vv0 += HSTEP * (c0a * hdv0[0] + c1a * hdv0[1] + c2a * hdv0[2] + c3a * hdv0[3]);
    vv1 += HSTEP * (c0a * hdv1[0] + c1a * hdv1[1] + c2a * hdv1[2] + c3a * hdv1[3]);

    // ---- store y_{it+1} : out[t][c][b][:] , half 0 -> x, half 1 -> v ----
    if (valid) {
      size_t off = ((size_t)((it + 1) * 2 + hh) * NB + (size_t)b) * 2;
      float2 w = hh ? make_float2(vv0, vv1) : make_float2(xx0, xx1);
      *reinterpret_cast<float2*>(out + off) = w;
    }
  }
}

extern "C" void kernel_launch(void* const* d_in, const int* in_sizes, int n_in,
                              void* d_out, int out_size, void* d_ws, size_t ws_size,
                              hipStream_t stream) {
  const float* x0    = (const float*)d_in[0];
  const float* v0    = (const float*)d_in[1];
  const float* beta  = (const float*)d_in[2];
  const float* ls    = (const float*)d_in[3];
  const float* alpha = (const float*)d_in[4];
  float* out = (float*)d_out;

  // 256 threads = 8 waves/block, 16 particles/wave -> 128 particles/block
  dim3 block(256);
  dim3 grid((NB + 127) / 128);
  geodesic_wmma_kernel<<<grid, block, 0, stream>>>(x0, v0, beta, ls, alpha, out);
}